// RBFLayer_42296837931709
// MI455X (gfx1250) — compile-verified
//
#include <hip/hip_runtime.h>

typedef float v2f __attribute__((ext_vector_type(2)));
typedef float v8f __attribute__((ext_vector_type(8)));

#define GAMMA  0.5f
#define M_DIM  4096
#define K_DIM  256
#define N_DIM  512

// ---------------------------------------------------------------------------
// W column norms: wn[n] = sum_d W[d,n]^2.  512 outputs, coalesced across n.
// ---------------------------------------------------------------------------
__global__ void rbf_wnorm_kernel(const float* __restrict__ W,
                                 float* __restrict__ wn) {
    int n = blockIdx.x * blockDim.x + threadIdx.x;   // 0..511
    float s = 0.0f;
#pragma unroll 8
    for (int d = 0; d < K_DIM; ++d) {
        float w = W[d * N_DIM + n];
        s = fmaf(w, w, s);
    }
    wn[n] = s;
}

// ---------------------------------------------------------------------------
// x row norms: one wave32 per row (256 floats = 64 float4 = 2 float4/lane),
// then a 5-step shuffle reduction.
// ---------------------------------------------------------------------------
__global__ void rbf_xnorm_kernel(const float* __restrict__ x,
                                 float* __restrict__ xn) {
    int wave = (blockIdx.x * blockDim.x + threadIdx.x) >> 5;  // row id
    int lane = threadIdx.x & 31;
    const float4* row = (const float4*)(x + (size_t)wave * K_DIM);
    float4 a = row[lane];
    float4 b = row[lane + 32];
    float s = a.x * a.x + a.y * a.y + a.z * a.z + a.w * a.w
            + b.x * b.x + b.y * b.y + b.z * b.z + b.w * b.w;
#pragma unroll
    for (int off = 16; off > 0; off >>= 1)
        s += __shfl_xor(s, off, 32);
    if (lane == 0) xn[wave] = s;
}

// ---------------------------------------------------------------------------
// Fused RBF GEMM:  out[m,n] = exp(2*g*(x.W)[m,n] - g*(xn[m] + wn[n]))
//
// Block = 256 threads = 8 waves arranged 4(M) x 2(N).
// Wave tile: 32(M) x 64(N) = 2 A-frags x 4 B-frags -> 8 accumulators.
// Each B-frag is reused by two WMMAs (upper/lower 16-row halves), giving
// 10 vmem : 8 wmma per K-step of 4 (vs 9:4 for a 16x64 tile).
// Block tile: 128 x 128.  Grid: (4096/128, 512/128) = (32, 4).
// K swept in steps of 4 with V_WMMA_F32_16X16X4_F32 (exact fp32).
// ---------------------------------------------------------------------------
__global__ void __launch_bounds__(256)
rbf_gemm_kernel(const float* __restrict__ x, const float* __restrict__ W,
                const float* __restrict__ xn, const float* __restrict__ wn,
                float* __restrict__ out) {
    const int lane = threadIdx.x & 31;
    const int wave = threadIdx.x >> 5;          // 0..7
    const int half = lane >> 4;                 // 0 or 1 (K sub-group)
    const int l16  = lane & 15;

    const int m0 = blockIdx.x * 128 + (wave & 3) * 32;   // wave's 32 rows
    const int n0 = blockIdx.y * 128 + (wave >> 2) * 64;  // wave's 64 cols

    v8f c[8] = {};   // c[t]  : rows m0..m0+15,  cols n0+16t
                     // c[t+4]: rows m0+16..+31, cols n0+16t

    // A-frag pointers: lane l16 -> row; half selects K pair {k,k+1} / {k+2,k+3}
    const float* aptr0 = x + (size_t)(m0 + l16) * K_DIM + half * 2;
    const float* aptr1 = aptr0 + 16 * K_DIM;
    // B-frag base: row (k + half*2) of W, column n0 + l16 (+ 16*t per tile)
    const float* bbase = W + n0 + l16;

    for (int k = 0; k < K_DIM; k += 4) {
        v2f a0 = *(const v2f*)(aptr0 + k);     // rows m0..m0+15
        v2f a1 = *(const v2f*)(aptr1 + k);     // rows m0+16..m0+31
        const float* bp = bbase + (size_t)(k + half * 2) * N_DIM;
#pragma unroll
        for (int t = 0; t < 4; ++t) {
            v2f b;
            b.x = bp[t * 16];
            b.y = bp[N_DIM + t * 16];
            c[t]     = __builtin_amdgcn_wmma_f32_16x16x4_f32(false, a0, false, b,
                                                             (short)0, c[t],     false, false);
            c[t + 4] = __builtin_amdgcn_wmma_f32_16x16x4_f32(false, a1, false, b,
                                                             (short)0, c[t + 4], false, false);
        }
    }

    // Row norms for this lane's 16 output rows
    // (accum VGPR r -> M = m0 + r + half*8 (+16 for the upper A group))
    float xr[16];
#pragma unroll
    for (int r = 0; r < 8; ++r) {
        xr[r]     = xn[m0 + r + half * 8];
        xr[r + 8] = xn[m0 + r + half * 8 + 16];
    }

#pragma unroll
    for (int t = 0; t < 4; ++t) {
        int n = n0 + t * 16 + l16;
        float wnn = wn[n];
#pragma unroll
        for (int r = 0; r < 8; ++r) {
            int m = m0 + r + half * 8;
            float e0 = __expf(fmaf(2.0f * GAMMA, c[t][r],
                                   -GAMMA * (xr[r] + wnn)));
            out[(size_t)m * N_DIM + n] = e0;
            float e1 = __expf(fmaf(2.0f * GAMMA, c[t + 4][r],
                                   -GAMMA * (xr[r + 8] + wnn)));
            out[(size_t)(m + 16) * N_DIM + n] = e1;
        }
    }
}

// ---------------------------------------------------------------------------
extern "C" void kernel_launch(void* const* d_in, const int* in_sizes, int n_in,
                              void* d_out, int out_size, void* d_ws, size_t ws_size,
                              hipStream_t stream) {
    const float* x = (const float*)d_in[0];   // 4096 x 256
    const float* W = (const float*)d_in[1];   // 256 x 512
    float* out = (float*)d_out;               // 4096 x 512

    float* wn = (float*)d_ws;                 // 512 floats
    float* xn = wn + N_DIM;                   // 4096 floats

    rbf_wnorm_kernel<<<dim3(N_DIM / 256), dim3(256), 0, stream>>>(W, wn);
    rbf_xnorm_kernel<<<dim3(M_DIM / 8),   dim3(256), 0, stream>>>(x, xn);

    dim3 grid(M_DIM / 128, N_DIM / 128);      // (32, 4)
    rbf_gemm_kernel<<<grid, dim3(256), 0, stream>>>(x, W, xn, wn, out);
}